// DeformMaxPool2d_77489799954449
// MI455X (gfx1250) — compile-verified
//
#include <hip/hip_runtime.h>
#include <hip/hip_bf16.h>
#include <stdint.h>

// DeformMaxPool2d: B=16, C=64, D=256, K=S=2 -> HO=128.
// out[b,c,p] = max_{k<4} x[b, c, gather_idx[p,k]]   (p in [0,16384), idx in [0,65536))
//
// MI455X plan (memory-bound: ~335MB compulsory traffic -> ~14us @ 23.3 TB/s):
//  - 1 workgroup per (b,c) image (1024 WGs x 1024 thr). Image = 256KB fits 320KB WGP LDS.
//  - Wave 0 issues one TDM tensor_load_to_lds (2D 256x256 f32 tile) streaming HBM->LDS,
//    waits s_wait_tensorcnt 0; meanwhile waves 1-31 prefetch their idx lines; barrier.
//  - Each thread: 4 iterations x 4 outputs: 128B contiguous idx load (b128s),
//    16 independent ds_load_b32 gathers, v_max tree, one global_store_b128.

#define NPIX   65536   // 256*256
#define NOUT   16384   // 128*128
#define NTHR   1024
#define OPT    4       // outputs per thread per iteration
#define NITER  (NOUT / (NTHR * OPT))   // = 4

typedef uint32_t u32;
typedef __attribute__((ext_vector_type(4))) u32 v4u;
typedef __attribute__((ext_vector_type(8))) int v8i;
typedef __attribute__((ext_vector_type(4))) int v4i;

__device__ __forceinline__ float max4(const longlong4& v, const float* lds) {
    const float a = lds[(u32)v.x];
    const float b = lds[(u32)v.y];
    const float c = lds[(u32)v.z];
    const float d = lds[(u32)v.w];
    return fmaxf(fmaxf(a, b), fmaxf(c, d));
}

__global__ __launch_bounds__(NTHR) void deform_maxpool_kernel(
    const float* __restrict__ x,
    const long long* __restrict__ gidx,
    float* __restrict__ out)
{
    extern __shared__ float lds[];   // 65536 floats = 256 KB

    const u32 bc  = blockIdx.x;      // image id in [0, 1024)
    const int tid = threadIdx.x;
    const float* __restrict__ img = x + (size_t)bc * NPIX;
    const longlong4* __restrict__ gi = (const longlong4*)gidx;  // [NOUT] x {4 int64}

    // ---- Wave 0: TDM DMA of the whole 256KB image into LDS --------------------
    if (tid < 32) {
        const uint64_t ga = (uint64_t)(uintptr_t)img;   // global addr of tile start
        const u32 lds_off = (u32)(uintptr_t)lds;        // LDS byte offset (low 32b of flat LDS addr)

        // D# group 0: [1:0]=count=1, [63:32]=lds_addr, [120:64]=global_addr, [127:126]=type=2
        v4u g0;
        g0.x = 1u;
        g0.y = lds_off;
        g0.z = (u32)(ga & 0xFFFFFFFFu);
        g0.w = (u32)((ga >> 32) & 0x01FFFFFFu) | (2u << 30);

        // D# group 1: data_size=4B; tensor 256x256, tile 256x256, dim0 stride 256
        v8i g1;
        g1[0] = (2 << 16);      // data_size=2 (4B)
        g1[1] = (256 << 16);    // tensor_dim0 lo16
        g1[2] = (256 << 16);    // tensor_dim1 lo16
        g1[3] = (256 << 16);    // tile_dim0
        g1[4] = 256;            // tile_dim1 (tile_dim2=0)
        g1[5] = 256;            // tensor_dim0_stride lo32
        g1[6] = 0;
        g1[7] = 0;

        v4i z4 = {0, 0, 0, 0};                 // groups 2/3: NULL (<=2D tensor)
        v8i z8 = {0, 0, 0, 0, 0, 0, 0, 0};

        __builtin_amdgcn_tensor_load_to_lds(g0, g1, z4, z4, z8, /*cpol=*/0);
        __builtin_amdgcn_s_wait_tensorcnt(0);
    } else {
        // While the DMA is in flight: warm the caches with exactly the idx lines
        // this thread will read (4 x 128B contiguous chunks, 128KB apart).
        #pragma unroll
        for (int it = 0; it < NITER; ++it)
            __builtin_prefetch(&gi[(it * NTHR + tid) * OPT], 0, 3);
    }
    __syncthreads();   // all 32 waves: LDS image now valid

    // ---- Gather + max-of-4 from LDS ------------------------------------------
    float* __restrict__ o = out + (size_t)bc * NOUT;

    #pragma unroll
    for (int it = 0; it < NITER; ++it) {
        const int p0 = (it * NTHR + tid) * OPT;   // 4 consecutive outputs / thread
        const longlong4 v0 = gi[p0 + 0];          // 128B contiguous per thread,
        const longlong4 v1 = gi[p0 + 1];          // consecutive threads adjacent
        const longlong4 v2 = gi[p0 + 2];
        const longlong4 v3 = gi[p0 + 3];
        float4 r;
        r.x = max4(v0, lds);
        r.y = max4(v1, lds);
        r.z = max4(v2, lds);
        r.w = max4(v3, lds);
        *(float4*)(o + p0) = r;                   // global_store_b128, coalesced
    }
}

extern "C" void kernel_launch(void* const* d_in, const int* in_sizes, int n_in,
                              void* d_out, int out_size, void* d_ws, size_t ws_size,
                              hipStream_t stream) {
    (void)in_sizes; (void)n_in; (void)d_ws; (void)ws_size; (void)out_size;
    const float*     x    = (const float*)d_in[0];
    const long long* gidx = (const long long*)d_in[1];
    float*           out  = (float*)d_out;

    // 1024 workgroups = one per (b,c) image; 256KB dynamic LDS each.
    dim3 grid(1024), block(NTHR);
    deform_maxpool_kernel<<<grid, block, NPIX * sizeof(float), stream>>>(x, gidx, out);
}